// TargetModel_65747359367630
// MI455X (gfx1250) — compile-verified
//
#include <hip/hip_runtime.h>
#include <math.h>

typedef __attribute__((ext_vector_type(2))) float v2f;
typedef __attribute__((ext_vector_type(8))) float v8f;

// ---------------- degree / normalization ----------------
__global__ void k_fill(float* __restrict__ p, float v, int n) {
  int i = blockIdx.x * blockDim.x + threadIdx.x;
  if (i < n) p[i] = v;
}

__global__ void k_deg_count(const int* __restrict__ dst, float* __restrict__ deg, int nE) {
  int i = blockIdx.x * blockDim.x + threadIdx.x;
  if (i < nE) unsafeAtomicAdd(&deg[dst[i]], 1.0f);
}

__global__ void k_rsqrt_inplace(float* __restrict__ p, int n) {
  int i = blockIdx.x * blockDim.x + threadIdx.x;
  if (i < n) p[i] = rsqrtf(p[i]);   // deg >= 1 always (self loop)
}

// ---------------- zero-pad B into ldp-wide buffer ----------------
__global__ void k_pad_B(const float* __restrict__ B, float* __restrict__ Bp,
                        int K, int N, int ldp) {
  int i = blockIdx.x * blockDim.x + threadIdx.x;
  int total = K * ldp;
  if (i >= total) return;
  int k = i / ldp;
  int c = i - k * ldp;
  Bp[i] = (c < N) ? B[(size_t)k * N + c] : 0.0f;
}

// ---------------- fp32 WMMA GEMM: C[M,N] = A[M,K] @ B[K,ldb-padded] ----------------
// One wave -> 16 rows x 64 cols (4 accumulators). Inner loop is branch-free:
// requires blockIdx.y*64 + 64 <= ldb (B zero-padded). Store guarded by col<N.
// blockDim.x = 128 (4 waves). grid = (ceil(M/16/4), ldb/64 groups actually used).
__global__ void k_wmma_gemm_f32(const float* __restrict__ A, const float* __restrict__ B,
                                float* __restrict__ C, int M, int N, int ldb, int K) {
  const int wavesPerBlock = blockDim.x >> 5;
  const int wave = threadIdx.x >> 5;
  const int lane = threadIdx.x & 31;
  const int lr   = lane & 15;
  const int kh   = lane >> 4;                        // K pair this lane feeds
  const int rowTile = blockIdx.x * wavesPerBlock + wave;
  const int numRowTiles = (M + 15) >> 4;
  if (rowTile >= numRowTiles) return;                // wave-uniform
  const int rowBase = rowTile * 16;
  const int colBase = blockIdx.y * 64 + lr;

  const int row  = rowBase + lr;
  const int rowL = row < M ? row : (M - 1);          // clamp loads
  const float* __restrict__ Arow = A + (size_t)rowL * K;

  v8f acc0 = {}, acc1 = {}, acc2 = {}, acc3 = {};
#pragma unroll 2
  for (int k = 0; k < K; k += 4) {
    const int ka = k + 2 * kh;
    v2f a;
    a.x = Arow[ka];
    a.y = Arow[ka + 1];
    const float* __restrict__ B0 = B + (size_t)ka * ldb + colBase;
    const float* __restrict__ B1 = B0 + ldb;
    v2f b0, b1, b2, b3;
    b0.x = B0[0];  b0.y = B1[0];
    b1.x = B0[16]; b1.y = B1[16];
    b2.x = B0[32]; b2.y = B1[32];
    b3.x = B0[48]; b3.y = B1[48];
    acc0 = __builtin_amdgcn_wmma_f32_16x16x4_f32(false, a, false, b0, (short)0, acc0, false, false);
    acc1 = __builtin_amdgcn_wmma_f32_16x16x4_f32(false, a, false, b1, (short)0, acc1, false, false);
    acc2 = __builtin_amdgcn_wmma_f32_16x16x4_f32(false, a, false, b2, (short)0, acc2, false, false);
    acc3 = __builtin_amdgcn_wmma_f32_16x16x4_f32(false, a, false, b3, (short)0, acc3, false, false);
  }

#pragma unroll
  for (int t = 0; t < 4; ++t) {
    v8f acc = (t == 0) ? acc0 : (t == 1) ? acc1 : (t == 2) ? acc2 : acc3;
    int col = colBase + 16 * t;
    if (col < N) {
#pragma unroll
      for (int i = 0; i < 8; ++i) {
        int r = rowBase + i + 8 * kh;
        if (r < M) C[(size_t)r * N + col] = acc[i];
      }
    }
  }
}

// ---------------- out[v,f] = h[v,f] * dinv[v]^2 + bias[f] ----------------
__global__ void k_selfloop_bias_init(const float* __restrict__ h, const float* __restrict__ dinv,
                                     const float* __restrict__ bias, float* __restrict__ out,
                                     int nNodes, int F) {
  long long i = (long long)blockIdx.x * blockDim.x + threadIdx.x;
  long long total = (long long)nNodes * F;
  if (i >= total) return;
  int v = (int)(i / F);
  int f = (int)(i - (long long)v * F);
  float di = dinv[v];
  out[i] = h[i] * di * di + bias[f];
}

// ---------------- per-edge scatter: out[dst] += h[src] * dinv[src]*dinv[dst] ----------
__global__ void k_scatter_edges(const float* __restrict__ h, const float* __restrict__ dinv,
                                const int* __restrict__ src, const int* __restrict__ dst,
                                float* __restrict__ out, int nE, int F) {
  int e = blockIdx.x;
  if (e >= nE) return;
  int s = src[e], d = dst[e];
  float norm = dinv[s] * dinv[d];
  const float* hs = h + (size_t)s * F;
  float* od = out + (size_t)d * F;
  for (int f = threadIdx.x; f < F; f += blockDim.x)
    unsafeAtomicAdd(&od[f], hs[f] * norm);
}

__global__ void k_relu(float* __restrict__ p, long long n) {
  long long i = (long long)blockIdx.x * blockDim.x + threadIdx.x;
  if (i < n) p[i] = fmaxf(p[i], 0.0f);
}

// ---------------- wave-per-row log_softmax (C <= 64) ----------------
__global__ void k_logsoftmax(float* __restrict__ out, int nNodes, int C) {
  int wavesPerBlock = blockDim.x >> 5;
  int w = blockIdx.x * wavesPerBlock + (threadIdx.x >> 5);
  if (w >= nNodes) return;
  int lane = threadIdx.x & 31;
  float* row = out + (size_t)w * C;
  float v0 = (lane      < C) ? row[lane]      : -INFINITY;
  float v1 = (lane + 32 < C) ? row[lane + 32] : -INFINITY;
  float m = fmaxf(v0, v1);
  for (int off = 16; off; off >>= 1) m = fmaxf(m, __shfl_xor(m, off));
  float s = ((lane < C) ? expf(v0 - m) : 0.0f) + ((lane + 32 < C) ? expf(v1 - m) : 0.0f);
  for (int off = 16; off; off >>= 1) s += __shfl_xor(s, off);
  float lse = m + logf(s);
  if (lane < C)      row[lane]      = v0 - lse;
  if (lane + 32 < C) row[lane + 32] = v1 - lse;
}

// ---------------- launcher ----------------
extern "C" void kernel_launch(void* const* d_in, const int* in_sizes, int n_in,
                              void* d_out, int out_size, void* d_ws, size_t ws_size,
                              hipStream_t stream) {
  const float* x  = (const float*)d_in[0];
  const int*   ei = (const int*)d_in[1];
  const float* W1 = (const float*)d_in[2];
  const float* b1 = (const float*)d_in[3];
  const float* W2 = (const float*)d_in[4];
  const float* b2 = (const float*)d_in[5];
  float* out = (float*)d_out;

  const int FEAT   = 128;
  const int HID    = in_sizes[3];          // 256
  const int NCLS   = in_sizes[5];          // 40
  const int NNODES = in_sizes[0] / FEAT;   // 100000
  const int NEDGES = in_sizes[1] / 2;      // 800000
  const int* src = ei;
  const int* dst = ei + NEDGES;

  // workspace (floats): [dinv @0 | W2pad @512KB | H @1MiB | O1], logits reuse H
  float* dinv  = (float*)d_ws;                        // NNODES
  float* W2pad = (float*)d_ws + (1 << 17);            // HID * 64
  float* H     = (float*)d_ws + (1 << 18);            // NNODES*HID
  float* O1    = H + (size_t)NNODES * HID;            // NNODES*HID
  float* L2    = H;                                   // NNODES*NCLS, reuse after ReLU

  const int numRowTiles = (NNODES + 15) / 16;         // 6250
  const int rowBlocks   = (numRowTiles + 3) / 4;

  // --- normalization: deg = 1 + #incoming, dinv = rsqrt(deg)
  k_fill<<<(NNODES + 255) / 256, 256, 0, stream>>>(dinv, 1.0f, NNODES);
  k_deg_count<<<(NEDGES + 255) / 256, 256, 0, stream>>>(dst, dinv, NEDGES);
  k_rsqrt_inplace<<<(NNODES + 255) / 256, 256, 0, stream>>>(dinv, NNODES);

  // --- layer 1: H = X @ W1 (WMMA fp32, N=256 = 4 col groups of 64)
  {
    dim3 grid(rowBlocks, HID / 64);
    k_wmma_gemm_f32<<<grid, 128, 0, stream>>>(x, W1, H, NNODES, HID, HID, FEAT);
  }
  long long tot1 = (long long)NNODES * HID;
  k_selfloop_bias_init<<<(unsigned)((tot1 + 255) / 256), 256, 0, stream>>>(H, dinv, b1, O1, NNODES, HID);
  k_scatter_edges<<<NEDGES, 256, 0, stream>>>(H, dinv, src, dst, O1, NEDGES, HID);
  k_relu<<<(unsigned)((tot1 + 255) / 256), 256, 0, stream>>>(O1, tot1);

  // --- layer 2: zero-pad W2 (HIDx40 -> HIDx64), then L2 = O1 @ W2pad
  k_pad_B<<<(HID * 64 + 255) / 256, 256, 0, stream>>>(W2, W2pad, HID, NCLS, 64);
  {
    dim3 grid(rowBlocks, 1);
    k_wmma_gemm_f32<<<grid, 128, 0, stream>>>(O1, W2pad, L2, NNODES, NCLS, 64, HID);
  }
  long long tot2 = (long long)NNODES * NCLS;
  k_selfloop_bias_init<<<(unsigned)((tot2 + 255) / 256), 256, 0, stream>>>(L2, dinv, b2, out, NNODES, NCLS);
  k_scatter_edges<<<NEDGES, 64, 0, stream>>>(L2, dinv, src, dst, out, NEDGES, NCLS);

  // --- log_softmax in place on d_out
  k_logsoftmax<<<(NNODES + 7) / 8, 256, 0, stream>>>(out, NNODES, NCLS);
}